// DeformableDynamicGather1D_27736898797673
// MI455X (gfx1250) — compile-verified
//
#include <hip/hip_runtime.h>
#include <math.h>

#define K_TAPS   5
#define TAU_F    2.0f
#define EPS_F    1e-8f
#define SLOPE_F  0.2f
#define C_DIM    256
#define LIN      4096
#define NPTS     8192
#define HDIM     64
#define INDIM    258   // C + 2
#define KPAD     260   // padded to multiple of 4
#define OUTD     12    // 1 + 1 + K + K

typedef float v2f __attribute__((ext_vector_type(2)));
typedef float v8f __attribute__((ext_vector_type(8)));

// Per-wave LDS slice (each wave owns a 16-point tile; no cross-wave sharing)
struct WaveSlice {
    float X[16][KPAD];   // input tile [point][in_dim padded]
    float Hb[16][HDIM];  // hidden activations
    float O[16][16];     // GEMM3 output tile (12 used)
    int   DI0[16][8];    // deform tap index 0 (padded rows)
    int   DI1[16][8];    // deform tap index 1
    float DT[16][8];     // deform lerp t
    float DW[16][8];     // normalized tap weight
    int   AI0[16];       // anchor i0
    int   AI1[16];       // anchor i1
    float AT[16];        // anchor lerp t
    float PX[16];        // x coord
    float PC[16];        // cell
};

__device__ __forceinline__ float leakyf(float x)    { return x >= 0.f ? x : SLOPE_F * x; }
__device__ __forceinline__ float softplusf(float x) { return x > 20.f ? x : log1pf(expf(x)); }
__device__ __forceinline__ float sigmoidf(float x)  { return 1.f / (1.f + expf(-x)); }
__device__ __forceinline__ int   imini(int a, int b){ return a < b ? a : b; }
__device__ __forceinline__ int   imaxi(int a, int b){ return a > b ? a : b; }

__global__ void ddg1d_fused(const float* __restrict__ feat,
                            const float* __restrict__ coords,
                            const float* __restrict__ cell,
                            const float* __restrict__ W1,
                            const float* __restrict__ b1,
                            const float* __restrict__ Wr,
                            const float* __restrict__ br,
                            const float* __restrict__ W3,
                            const float* __restrict__ b3,
                            float* __restrict__ out)
{
    extern __shared__ char smem_raw[];
    const int lane = threadIdx.x & 31;
    const int wave = threadIdx.x >> 5;
    const int half = lane >> 4;     // 0: lanes 0-15, 1: lanes 16-31
    const int m    = lane & 15;     // row within 16x16 tile / N column index

    WaveSlice* ws = reinterpret_cast<WaveSlice*>(smem_raw) + wave;

    const int tile = blockIdx.x * 4 + wave;   // 4 waves per block, 2048 tiles
    const int gid0 = tile * 16;               // first flat point id (b*N + n)
    const int b    = gid0 / NPTS;
    const int feat_base = b * C_DIM * LIN;

    // ---------------- per-point coords + anchor sample setup ----------------
    if (lane < 16) {
        const int gid = gid0 + lane;
        const float x  = coords[gid];
        const float cl = cell[gid];
        ws->PX[lane] = x;
        ws->PC[lane] = cl;
        float ix = (x + 1.f) * 0.5f * (float)(LIN - 1);
        ix = fminf(fmaxf(ix, 0.f), (float)(LIN - 1));
        const float x0 = floorf(ix);
        int i0 = imini(imaxi((int)x0, 0), LIN - 1);
        int i1 = imini(i0 + 1, LIN - 1);
        ws->AI0[lane] = i0;
        ws->AI1[lane] = i1;
        ws->AT[lane]  = ix - x0;
        ws->X[lane][256] = x;
        ws->X[lane][257] = cl;
        ws->X[lane][258] = 0.f;
        ws->X[lane][259] = 0.f;
    }
    __syncthreads();

    // ---------------- anchor gather into X tile ----------------
    {
        const int   i0 = ws->AI0[m];
        const int   i1 = ws->AI1[m];
        const float t  = ws->AT[m];
        const float* fb = feat + feat_base;
        #pragma unroll 4
        for (int j = 0; j < 128; ++j) {
            const int c = 2 * j + half;       // lanes cover 16 pts x 2 channels
            const float f0 = fb[c * LIN + i0];
            const float f1 = fb[c * LIN + i1];
            ws->X[m][c] = f0 + (f1 - f0) * t;
        }
    }
    __syncthreads();

    // ---------------- GEMM1: X[16x260] @ W1[258x64] (f32 WMMA) ----------------
    v8f acc[4];
    #pragma unroll
    for (int nt = 0; nt < 4; ++nt)
        acc[nt] = (v8f){0.f, 0.f, 0.f, 0.f, 0.f, 0.f, 0.f, 0.f};

    // main body: rows 0..255, all in-bounds -> unconditional loads, no exec games
    #pragma unroll 2
    for (int k = 0; k < 256; k += 4) {
        const int r0 = k + 2 * half;
        v2f a;
        a.x = ws->X[m][r0];
        a.y = ws->X[m][r0 + 1];
        #pragma unroll
        for (int nt = 0; nt < 4; ++nt) {
            const int col = nt * 16 + m;
            v2f bf;
            bf.x = W1[ r0      * HDIM + col];
            bf.y = W1[(r0 + 1) * HDIM + col];
            acc[nt] = __builtin_amdgcn_wmma_f32_16x16x4_f32(
                false, a, false, bf, (short)0, acc[nt], false, false);
        }
    }
    // peeled final step: rows 256..259 (258/259 are zero pad) -> clamp + select
    {
        const int r0 = 256 + 2 * half;                  // 256 (half=0) or 258 (half=1)
        const int rc0 = imini(r0,     INDIM - 1);
        const int rc1 = imini(r0 + 1, INDIM - 1);
        const float m0 = (r0     < INDIM) ? 1.f : 0.f;  // v_cndmask, no branch
        const float m1 = (r0 + 1 < INDIM) ? 1.f : 0.f;
        v2f a;
        a.x = ws->X[m][r0];
        a.y = ws->X[m][r0 + 1];
        #pragma unroll
        for (int nt = 0; nt < 4; ++nt) {
            const int col = nt * 16 + m;
            v2f bf;
            bf.x = W1[rc0 * HDIM + col] * m0;
            bf.y = W1[rc1 * HDIM + col] * m1;
            acc[nt] = __builtin_amdgcn_wmma_f32_16x16x4_f32(
                false, a, false, bf, (short)0, acc[nt], false, false);
        }
    }

    // bias + leaky, stash h both in registers (residual seed) and LDS (A reads)
    float h1[4][8];
    #pragma unroll
    for (int nt = 0; nt < 4; ++nt) {
        const int n  = nt * 16 + m;
        const float bb = b1[n];
        #pragma unroll
        for (int r = 0; r < 8; ++r) {
            float v = leakyf(acc[nt][r] + bb);
            h1[nt][r] = v;
            ws->Hb[r + 8 * half][n] = v;
        }
    }
    __syncthreads();

    // ---------------- GEMM2: h @ Wr, residual seeded into C ----------------
    v8f acc2[4];
    #pragma unroll
    for (int nt = 0; nt < 4; ++nt) {
        const int n = nt * 16 + m;
        const float bb = br[n];
        #pragma unroll
        for (int r = 0; r < 8; ++r)
            acc2[nt][r] = h1[nt][r] + bb;   // C = h + br  =>  D = h@Wr + h + br
    }
    #pragma unroll 4
    for (int k = 0; k < HDIM; k += 4) {
        const int r0 = k + 2 * half;
        v2f a;
        a.x = ws->Hb[m][r0];
        a.y = ws->Hb[m][r0 + 1];
        #pragma unroll
        for (int nt = 0; nt < 4; ++nt) {
            const int col = nt * 16 + m;
            v2f bf;
            bf.x = Wr[ r0      * HDIM + col];
            bf.y = Wr[(r0 + 1) * HDIM + col];
            acc2[nt] = __builtin_amdgcn_wmma_f32_16x16x4_f32(
                false, a, false, bf, (short)0, acc2[nt], false, false);
        }
    }
    __syncthreads();   // all GEMM2 A-reads done before Hb overwrite
    #pragma unroll
    for (int nt = 0; nt < 4; ++nt) {
        const int n = nt * 16 + m;
        #pragma unroll
        for (int r = 0; r < 8; ++r)
            ws->Hb[r + 8 * half][n] = leakyf(acc2[nt][r]);
    }
    __syncthreads();

    // ---------------- GEMM3: h2 @ W3[64x12] (one padded N-tile) ----------------
    // lane-divergent column bound handled with clamped address + 0/1 mask (VALU only)
    const int   mc    = imini(m, OUTD - 1);
    const float mmask = (m < OUTD) ? 1.f : 0.f;
    v8f acc3 = (v8f){0.f, 0.f, 0.f, 0.f, 0.f, 0.f, 0.f, 0.f};
    #pragma unroll 4
    for (int k = 0; k < HDIM; k += 4) {
        const int r0 = k + 2 * half;
        v2f a;
        a.x = ws->Hb[m][r0];
        a.y = ws->Hb[m][r0 + 1];
        v2f bf;
        bf.x = W3[ r0      * OUTD + mc] * mmask;
        bf.y = W3[(r0 + 1) * OUTD + mc] * mmask;
        acc3 = __builtin_amdgcn_wmma_f32_16x16x4_f32(
            false, a, false, bf, (short)0, acc3, false, false);
    }
    if (m < OUTD) {
        const float bb = b3[m];
        #pragma unroll
        for (int r = 0; r < 8; ++r)
            ws->O[r + 8 * half][m] = acc3[r] + bb;
    }
    __syncthreads();

    // ---------------- per-point head: offsets + normalized tap weights ----------------
    if (lane < 16) {
        const int p = lane;
        const float r_v = fminf(fmaxf(softplusf(ws->O[p][0]) + 0.3f, 0.3f), 2.0f);
        const float s_v = fminf(fmaxf(softplusf(ws->O[p][1]) + 0.5f, 0.5f), 3.0f);
        const float denom = (s_v * TAU_F) * (s_v * TAU_F) + EPS_F;
        const float x = ws->PX[p];
        const float base[K_TAPS] = {-2.f, -1.f, 0.f, 1.f, 2.f};
        const float scale_x = 2.0f / (float)(LIN - 1);
        float wv[K_TAPS];
        float wsum = 0.f;
        #pragma unroll
        for (int k = 0; k < K_TAPS; ++k) {
            const float res  = tanhf(ws->O[p][2 + k]) * 0.5f;
            const float off  = r_v * base[k] + res;
            const float gate = ws->O[p][2 + K_TAPS + k];
            const float w = expf(-0.5f * off * off / denom) * sigmoidf(gate);
            wv[k] = w;
            wsum += w;
            const float dx = x + off * scale_x;
            float ix = (dx + 1.f) * 0.5f * (float)(LIN - 1);
            ix = fminf(fmaxf(ix, 0.f), (float)(LIN - 1));
            const float x0 = floorf(ix);
            int i0 = imini(imaxi((int)x0, 0), LIN - 1);
            int i1 = imini(i0 + 1, LIN - 1);
            ws->DI0[p][k] = i0;
            ws->DI1[p][k] = i1;
            ws->DT[p][k]  = ix - x0;
        }
        const float inv = 1.f / (wsum + EPS_F);
        #pragma unroll
        for (int k = 0; k < K_TAPS; ++k)
            ws->DW[p][k] = wv[k] * inv;
    }
    __syncthreads();

    // ---------------- 5-tap weighted gather, channels across lanes ----------------
    for (int p = 0; p < 16; ++p) {
        const int gid = gid0 + p;
        float* op = out + (size_t)gid * C_DIM;
        int i0[K_TAPS], i1[K_TAPS];
        float t[K_TAPS], w[K_TAPS];
        #pragma unroll
        for (int k = 0; k < K_TAPS; ++k) {
            i0[k] = ws->DI0[p][k];
            i1[k] = ws->DI1[p][k];
            t[k]  = ws->DT[p][k];
            w[k]  = ws->DW[p][k];
        }
        #pragma unroll 2
        for (int g = 0; g < 8; ++g) {
            const int c = g * 32 + lane;
            const float* fb = feat + feat_base + c * LIN;
            float accv = 0.f;
            #pragma unroll
            for (int k = 0; k < K_TAPS; ++k) {
                const float f0 = fb[i0[k]];
                const float f1 = fb[i1[k]];
                accv += w[k] * (f0 + (f1 - f0) * t[k]);
            }
            op[c] = accv;   // coalesced 128B store per wave
        }
    }
}

extern "C" void kernel_launch(void* const* d_in, const int* in_sizes, int n_in,
                              void* d_out, int out_size, void* d_ws, size_t ws_size,
                              hipStream_t stream) {
    const float* feat   = (const float*)d_in[0];
    const float* coords = (const float*)d_in[1];
    const float* cellp  = (const float*)d_in[2];
    const float* W1     = (const float*)d_in[3];
    const float* b1     = (const float*)d_in[4];
    const float* Wr     = (const float*)d_in[5];
    const float* br     = (const float*)d_in[6];
    const float* W3     = (const float*)d_in[7];
    const float* b3     = (const float*)d_in[8];
    float* out = (float*)d_out;

    // 2048 tiles of 16 points; 4 waves (128 threads) per block -> 512 blocks
    const dim3 grid(512), block(128);
    const size_t smem = 4 * sizeof(WaveSlice);   // ~96 KB of the 320 KB WGP LDS
    hipLaunchKernelGGL(ddg1d_fused, grid, block, smem, stream,
                       feat, coords, cellp, W1, b1, Wr, br, W3, b3, out);
}